// SparseConv2D_12927851561353
// MI455X (gfx1250) — compile-verified
//
#include <hip/hip_runtime.h>
#include <hip/hip_bf16.h>
#include <cstddef>

typedef __attribute__((ext_vector_type(2))) float v2f;
typedef __attribute__((ext_vector_type(8))) float v8f;

#define H_IN   1024
#define W_IN   1024
#define FRAMES 32
#define TAPS   25
#define PH     1028          // padded height (2 halo each side)
#define PW     1032          // padded row stride (1028 cols + 4 align slack)

// ---------------------------------------------------------------------------
// Pre-pass: build zero-padded input (PH x PW) in workspace.
// pad[r][c] = in[r-2][c-2] when inside, else 0.
// ---------------------------------------------------------------------------
__global__ void sconv_pad_kernel(const float* __restrict__ in,
                                 float* __restrict__ pad) {
    int i = blockIdx.x * blockDim.x + threadIdx.x;
    if (i >= PH * PW) return;
    int r = i / PW;
    int c = i - r * PW;
    float v = 0.0f;
    if (r >= 2 && r < 2 + H_IN && c >= 2 && c < 2 + W_IN)
        v = in[(r - 2) * W_IN + (c - 2)];
    pad[i] = v;
}

// ---------------------------------------------------------------------------
// Main kernel: C[32, HW] = W[32,25] x U[25, HW] via V_WMMA_F32_16X16X4_F32.
// Grid: 2 blocks per image row (2048 blocks), 256 threads (8 waves).
// Each wave: 4 N-tiles of 16 pixels x 2 M-tiles of 16 frames.
// K=25 padded to 28 -> 7 WMMA K-steps; pad taps get zero A-weights
// (loads are clamped to valid addresses; zeroing is a branch-free cndmask).
// ---------------------------------------------------------------------------
__global__ __launch_bounds__(256) void sconv_wmma_kernel(
    const float* __restrict__ pad,   // PH x PW padded input
    const float* __restrict__ wk,    // FRAMES x TAPS weights
    const int*   __restrict__ idx,   // TAPS x 2 (dx, dy)
    float*       __restrict__ out)   // FRAMES x H x W
{
    const int lane = threadIdx.x & 31;
    const int wv   = threadIdx.x >> 5;          // 0..7
    const int bid  = blockIdx.x;
    const int h    = bid >> 1;                  // image row
    const int half = bid & 1;
    const int wbase = half * 512 + wv * 64;     // first pixel column of wave
    const int lm = lane & 15;                   // N (and M-row) within tile
    const int hi = lane >> 4;                   // K-half selector

    // -------- A tiles (weights) + per-K-group B element offsets --------
    v2f a[2][7];
    int off0[7], off1[7];
#pragma unroll
    for (int g = 0; g < 7; ++g) {
        int k0 = 4 * g + 2 * hi;
        int k1 = k0 + 1;
        int k0c = (k0 < TAPS) ? k0 : (TAPS - 1);   // clamp pad taps to a
        int k1c = (k1 < TAPS) ? k1 : (TAPS - 1);   // valid (finite) address
#pragma unroll
        for (int m = 0; m < 2; ++m) {
            // Branch-free: always load from clamped (valid) address, then
            // zero the K-padding weights with a select (v_cndmask).
            float ax = wk[(m * 16 + lm) * TAPS + k0c];
            float ay = wk[(m * 16 + lm) * TAPS + k1c];
            a[m][g].x = (k0 < TAPS) ? ax : 0.0f;
            a[m][g].y = (k1 < TAPS) ? ay : 0.0f;
        }
        int dx0 = idx[2 * k0c], dy0 = idx[2 * k0c + 1];
        int dx1 = idx[2 * k1c], dy1 = idx[2 * k1c + 1];
        // element offset inside padded image, relative to (h, wbase)
        off0[g] = (dy0 + 2) * PW + (dx0 + 2) + lm;
        off1[g] = (dy1 + 2) * PW + (dx1 + 2) + lm;
    }

    const float* rowp = pad + (size_t)h * PW + wbase;

    v8f acc[4][2];
#pragma unroll
    for (int t = 0; t < 4; ++t)
#pragma unroll
        for (int m = 0; m < 2; ++m)
            acc[t][m] = (v8f){0.f, 0.f, 0.f, 0.f, 0.f, 0.f, 0.f, 0.f};

    // ------------------------------ GEMM loop ------------------------------
#pragma unroll
    for (int g = 0; g < 7; ++g) {
#pragma unroll
        for (int t = 0; t < 4; ++t) {
            v2f b;
            b.x = rowp[t * 16 + off0[g]];
            b.y = rowp[t * 16 + off1[g]];
            acc[t][0] = __builtin_amdgcn_wmma_f32_16x16x4_f32(
                false, a[0][g], false, b, (short)0, acc[t][0], false, false);
            acc[t][1] = __builtin_amdgcn_wmma_f32_16x16x4_f32(
                false, a[1][g], false, b, (short)0, acc[t][1], false, false);
        }
    }

    // ----------------------------- store (NT) ------------------------------
    // C/D layout: VGPR j, lane L -> M = j + 8*(L>>4), N = L&15.
#pragma unroll
    for (int t = 0; t < 4; ++t) {
#pragma unroll
        for (int m = 0; m < 2; ++m) {
#pragma unroll
            for (int j = 0; j < 8; ++j) {
                int f = m * 16 + j + 8 * hi;
                size_t o = ((size_t)f << 20) + (size_t)h * W_IN + wbase + t * 16 + lm;
                __builtin_nontemporal_store(acc[t][m][j], out + o);
            }
        }
    }
}

// ---------------------------------------------------------------------------
// Fallback (only if workspace is too small for the padded image): scalar,
// bounds-checked, one thread per pixel, all 32 frames accumulated in regs.
// ---------------------------------------------------------------------------
__global__ __launch_bounds__(256) void sconv_fallback_kernel(
    const float* __restrict__ in,
    const float* __restrict__ wk,
    const int*   __restrict__ idx,
    float*       __restrict__ out)
{
    int p = blockIdx.x * blockDim.x + threadIdx.x;
    if (p >= H_IN * W_IN) return;
    int h = p >> 10;
    int w = p & 1023;
    float acc[FRAMES];
#pragma unroll
    for (int f = 0; f < FRAMES; ++f) acc[f] = 0.0f;
    for (int k = 0; k < TAPS; ++k) {
        int dx = idx[2 * k], dy = idx[2 * k + 1];
        int hh = h + dy, ww = w + dx;
        float v = 0.0f;
        if (hh >= 0 && hh < H_IN && ww >= 0 && ww < W_IN)
            v = in[hh * W_IN + ww];
#pragma unroll
        for (int f = 0; f < FRAMES; ++f)
            acc[f] += wk[f * TAPS + k] * v;
    }
#pragma unroll
    for (int f = 0; f < FRAMES; ++f)
        out[((size_t)f << 20) + p] = acc[f];
}

extern "C" void kernel_launch(void* const* d_in, const int* in_sizes, int n_in,
                              void* d_out, int out_size, void* d_ws, size_t ws_size,
                              hipStream_t stream) {
    const float* in  = (const float*)d_in[0];   // 1024x1024 f32
    const float* wk  = (const float*)d_in[1];   // 32x25 f32
    const int*   idx = (const int*)d_in[2];     // 25x2 i32
    float* out = (float*)d_out;                 // 32x1024x1024 f32

    const size_t pad_bytes = (size_t)PH * PW * sizeof(float);
    if (ws_size >= pad_bytes) {
        float* pad = (float*)d_ws;
        int n = PH * PW;
        sconv_pad_kernel<<<(n + 255) / 256, 256, 0, stream>>>(in, pad);
        // 2 blocks per row x 1024 rows
        sconv_wmma_kernel<<<2 * H_IN, 256, 0, stream>>>(pad, wk, idx, out);
    } else {
        int n = H_IN * W_IN;
        sconv_fallback_kernel<<<(n + 255) / 256, 256, 0, stream>>>(in, wk, idx, out);
    }
}